// TemporalHybrid_34402688041056
// MI455X (gfx1250) — compile-verified
//
#include <hip/hip_runtime.h>
#include <stdint.h>

// ---------------------------------------------------------------------------
// TemporalHybrid (S4 + TCN) forward for MI455X / gfx1250, all-FP32 WMMA.
//
// Shapes: B=32, L=1024, D_IN=64, H=N=256.  Total matmul work ~17 GFLOP,
// total memory traffic ~200MB -> memory/latency bound at MI455X scale, so we
// keep everything in f32 and use V_WMMA_F32_16X16X4_F32 for exactness.
//
// Pipeline (all on `stream`, scratch in d_ws, ~97MB):
//   1. k_gemm_bias<64>  : U = x @ enc_w + enc_b                (WMMA)
//   2. k_ssm_kernel     : K[h,l] = 2*Re(CB * exp(dtA*l))       (VALU trans)
//   3. k_conv           : Yt[h,l,b] = causal Toeplitz matmul   (WMMA +
//                         global_load_async_to_lds_b128 staging)
//   4. k_ln1            : blk = LN(gelu(y + u*D) + u)
//   5. k_gemm_bias<256> : out = blk @ dec_w + dec_b            (WMMA)
//   6. k_tcn            : out += dilated conv1d (3 shifted GEMMs, WMMA)
//   7. k_lnf            : out = LN(out) in-place
// ---------------------------------------------------------------------------

typedef float v2f __attribute__((ext_vector_type(2)));
typedef float v8f __attribute__((ext_vector_type(8)));

#define B_SZ    32
#define L_SEQ   1024
#define DIN     64
#define H_DIM   256
#define N_STATE 256
#define M_ROWS  (B_SZ * L_SEQ)   // 32768

__device__ __forceinline__ v8f wmma_f32(v2f a, v2f b, v8f c) {
  // D = A(16x4) * B(4x16) + C(16x16), fp32, wave32.
  // 8 args: (neg_a, A, neg_b, B, c_mod, C, reuse_a, reuse_b)
  return __builtin_amdgcn_wmma_f32_16x16x4_f32(false, a, false, b, (short)0, c,
                                               false, false);
}

// ---------------------------------------------------------------------------
// Generic GEMM: Out[M_ROWS, 256] = A[M_ROWS, K] @ W[K, 256] + bias.
// One wave per 16x16 output tile; 8 waves per block.
// A lane layout: (m = lane&15, k = 2*laneHi + vgpr)
// B lane layout: (k = 2*laneHi + vgpr, n = lane&15)
// ---------------------------------------------------------------------------
template <int K>
__global__ void k_gemm_bias(const float* __restrict__ A,
                            const float* __restrict__ W,
                            const float* __restrict__ bias,
                            float* __restrict__ Out) {
  const int wave = threadIdx.x >> 5;
  const int lane = threadIdx.x & 31;
  const int lh   = lane >> 4;
  const int ml   = lane & 15;
  const int tile = blockIdx.x * 8 + wave;   // 32768 tiles total
  const int tn   = tile & 15;               // 256/16 = 16 tiles in N
  const int tm   = tile >> 4;               // 2048 tiles in M
  const int row  = tm * 16 + ml;
  const int col  = tn * 16 + ml;

  v8f acc = {0.f, 0.f, 0.f, 0.f, 0.f, 0.f, 0.f, 0.f};
#pragma unroll 8
  for (int kk = 0; kk < K / 4; ++kk) {
    const int kb = kk * 4 + lh * 2;
    v2f a = *(const v2f*)(A + row * K + kb);   // 8B aligned: kb even
    v2f b;
    b.x = W[(kb + 0) * H_DIM + col];
    b.y = W[(kb + 1) * H_DIM + col];
    acc = wmma_f32(a, b, acc);
  }
  const float bv = bias[col];
#pragma unroll
  for (int r = 0; r < 8; ++r) {
    const int orow = tm * 16 + r + 8 * lh;   // C layout: M = r + 8*laneHi
    Out[orow * H_DIM + col] = acc[r] + bv;
  }
}

// ---------------------------------------------------------------------------
// SSM conv kernel K[h,l] = sum_n e^{a_n l} (cbr_n cos(b_n l) - cbi_n sin(b_n l))
//   a = -dt*exp(log_A_real), b = dt*A_imag, cb = 2*dt*(C_re+iC_im)(B_re+iB_im)
// One block per (h, 256 l-values); per-h params staged once in LDS.
// ---------------------------------------------------------------------------
__global__ void k_ssm_kernel(const float* __restrict__ lAr,
                             const float* __restrict__ Aim,
                             const float* __restrict__ Bre,
                             const float* __restrict__ Bim,
                             const float* __restrict__ Cre,
                             const float* __restrict__ Cim,
                             const float* __restrict__ ldt,
                             float* __restrict__ Kg) {
  __shared__ float sa[N_STATE], sb[N_STATE], scr[N_STATE], sci[N_STATE];
  const int h = blockIdx.y;
  const int t = threadIdx.x;
  const float dt = __expf(ldt[h]);
  {
    const int i  = h * N_STATE + t;
    const float br = Bre[i], bi = Bim[i], cr = Cre[i], ci = Cim[i];
    sa[t]  = -__expf(lAr[i]) * dt;
    sb[t]  = Aim[i] * dt;
    scr[t] = 2.f * dt * (cr * br - ci * bi);
    sci[t] = 2.f * dt * (cr * bi + ci * br);
  }
  __syncthreads();
  const int   l  = blockIdx.x * blockDim.x + t;
  const float fl = (float)l;
  float acc = 0.f;
#pragma unroll 4
  for (int n = 0; n < N_STATE; ++n) {
    const float e = __expf(sa[n] * fl);
    float s, c;
    __sincosf(sb[n] * fl, &s, &c);
    acc += e * (scr[n] * c - sci[n] * s);
  }
  Kg[h * L_SEQ + l] = acc;
}

// ---------------------------------------------------------------------------
// Causal convolution as lower-triangular Toeplitz WMMA matmul, per channel h:
//   Yt[h, l, b] = sum_{j<=l} K[h, l-j] * U[(b,j), h]
// Block = 4 waves, all same h.  K[h,:] (4KB) is staged into LDS with the
// CDNA5 async data-mover path (global_load_async_to_lds_b128 + ASYNCcnt),
// with a 16-float zero pad in front so the causal/diagonal masking is a
// plain LDS indexed read (no conditionals, EXEC stays all-ones around WMMA).
// The j-tile loop bound is forced into an SGPR (wave-uniform) so the loop is
// a scalar branch, not a lane-masked loop.
// Output stored [H,L,B] so stores coalesce.
// ---------------------------------------------------------------------------
__global__ void k_conv(const float* __restrict__ U,
                       const float* __restrict__ Kg,
                       float* __restrict__ Yt) {
  __shared__ float sKpad[16 + L_SEQ];
  const int h   = blockIdx.y;
  const int tid = threadIdx.x;    // 128

  // Async-stage K[h, 0..1023] into sKpad[16..]: 2 passes x 128 lanes x 16B.
  {
    const uint32_t lds_base =
        (uint32_t)(uintptr_t)(&sKpad[16]);          // LDS byte offset
    const uint64_t gbase = (uint64_t)(uintptr_t)(Kg + h * L_SEQ);
#pragma unroll
    for (int p = 0; p < 2; ++p) {
      const uint32_t off  = (uint32_t)(tid * 16 + p * 2048);
      const uint32_t ldsa = lds_base + off;
      asm volatile("global_load_async_to_lds_b128 %0, %1, %2"
                   :
                   : "v"(ldsa), "v"(off), "s"(gbase)
                   : "memory");
    }
  }
  if (tid < 16) sKpad[tid] = 0.f;                   // causal zero pad
  asm volatile("s_wait_asynccnt 0x0" ::: "memory");
  __syncthreads();
  const float* sK = &sKpad[16];

  const int wave = tid >> 5, lane = tid & 31;
  const int lh = lane >> 4, ml = lane & 15;
  const int lt = __builtin_amdgcn_readfirstlane(blockIdx.x * 2 + (wave >> 1));
  const int bt = wave & 1;
  const int l0 = lt * 16, b0 = bt * 16;

  v8f acc = {0.f, 0.f, 0.f, 0.f, 0.f, 0.f, 0.f, 0.f};
  for (int jt = 0; jt <= lt; ++jt) {
    const int j0 = jt * 16;
    const int d0 = l0 - j0;
    if (jt < lt)  // prefetch next j-tile of U for this lane's b-row
      __builtin_prefetch(U + ((b0 + ml) * L_SEQ + j0 + 16) * H_DIM + h, 0, 1);
#pragma unroll
    for (int q = 0; q < 4; ++q) {
      const int kk = 4 * q + 2 * lh;
      const int ia = d0 + ml - kk;          // Toeplitz index l-j; >= -16
      v2f a, b;
      a.x = sK[ia];                          // pad makes negatives read 0
      a.y = sK[ia - 1];
      const float* up = U + ((b0 + ml) * L_SEQ + (j0 + kk)) * H_DIM + h;
      b.x = up[0];
      b.y = up[H_DIM];
      acc = wmma_f32(a, b, acc);
    }
  }
#pragma unroll
  for (int r = 0; r < 8; ++r) {
    const int l = l0 + r + 8 * lh;
    Yt[h * (L_SEQ * B_SZ) + l * B_SZ + b0 + ml] = acc[r];
  }
}

// ---------------------------------------------------------------------------
// y = Yt + u*D; g = gelu_tanh(y); blk = LayerNorm_H(g + u) * ln1_g + ln1_b
// One block per (b,l) row; 256 threads = one h each; LDS tree reduction.
// ---------------------------------------------------------------------------
__global__ void k_ln1(const float* __restrict__ U,
                      const float* __restrict__ Yt,
                      const float* __restrict__ Dp,
                      const float* __restrict__ g,
                      const float* __restrict__ bb,
                      float* __restrict__ Blk) {
  __shared__ float ssum[H_DIM], ssq[H_DIM];
  const int m = blockIdx.x, t = threadIdx.x;
  const int b = m >> 10, l = m & 1023;
  const float uv = U[m * H_DIM + t];
  const float y  = Yt[t * (L_SEQ * B_SZ) + l * B_SZ + b] + uv * Dp[t];
  const float gl =
      0.5f * y * (1.f + tanhf(0.7978845608028654f * (y + 0.044715f * y * y * y)));
  const float r = gl + uv;
  ssum[t] = r;
  ssq[t]  = r * r;
  __syncthreads();
  for (int off = H_DIM / 2; off > 0; off >>= 1) {
    if (t < off) {
      ssum[t] += ssum[t + off];
      ssq[t]  += ssq[t + off];
    }
    __syncthreads();
  }
  const float mean = ssum[0] * (1.f / H_DIM);
  const float var  = ssq[0] * (1.f / H_DIM) - mean * mean;
  Blk[m * H_DIM + t] = (r - mean) * rsqrtf(var + 1e-5f) * g[t] + bb[t];
}

// ---------------------------------------------------------------------------
// TCN: Conv1d(D_IN->H, k=3, dilation=2, pad=2) as 3 shifted GEMMs over K=64.
// Zero-padding folded into the A fragment via clamped (always-safe) loads
// multiplied by a 0/1 mask -> v_cndmask, no EXEC manipulation near WMMA.
// Accumulates into d_out (which already holds the decoder result).
// ---------------------------------------------------------------------------
__global__ void k_tcn(const float* __restrict__ X,
                      const float* __restrict__ Wt,   // [H, DIN, 3]
                      const float* __restrict__ tb,
                      float* __restrict__ Out) {
  const int wave = threadIdx.x >> 5, lane = threadIdx.x & 31;
  const int lh = lane >> 4, ml = lane & 15;
  const int tile = blockIdx.x * 8 + wave;
  const int tn = tile & 15, tm = tile >> 4;
  const int bidx  = (tm * 16) >> 10;     // 16 | 1024 so tile stays in one batch
  const int lbase = (tm * 16) & 1023;
  const int col   = tn * 16 + ml;        // h
  const int lrow  = lbase + ml;

  v8f acc = {0.f, 0.f, 0.f, 0.f, 0.f, 0.f, 0.f, 0.f};
#pragma unroll
  for (int t = 0; t < 3; ++t) {
    const int j  = lrow + 2 * t - 2;     // dilation 2, padding 2
    int jc = j < 0 ? 0 : j;              // clamp -> load is always in-bounds
    jc = jc > (L_SEQ - 1) ? (L_SEQ - 1) : jc;
    const float msk = (j == jc) ? 1.f : 0.f;
    const float* xr = X + (bidx * L_SEQ + jc) * DIN;
#pragma unroll
    for (int cc = 0; cc < DIN / 4; ++cc) {
      const int c0 = cc * 4 + 2 * lh;
      v2f a = *(const v2f*)(xr + c0);
      a.x *= msk;
      a.y *= msk;
      v2f b;
      b.x = Wt[col * (DIN * 3) + (c0 + 0) * 3 + t];
      b.y = Wt[col * (DIN * 3) + (c0 + 1) * 3 + t];
      acc = wmma_f32(a, b, acc);
    }
  }
  const float bv = tb[col];
#pragma unroll
  for (int r = 0; r < 8; ++r) {
    const int orow = tm * 16 + r + 8 * lh;
    const int idx  = orow * H_DIM + col;
    Out[idx] = Out[idx] + acc[r] + bv;
  }
}

// ---------------------------------------------------------------------------
// Final LayerNorm over H, in place on d_out.
// ---------------------------------------------------------------------------
__global__ void k_lnf(float* __restrict__ Out,
                      const float* __restrict__ g,
                      const float* __restrict__ bb) {
  __shared__ float ssum[H_DIM], ssq[H_DIM];
  const int m = blockIdx.x, t = threadIdx.x;
  const float r = Out[m * H_DIM + t];
  ssum[t] = r;
  ssq[t]  = r * r;
  __syncthreads();
  for (int off = H_DIM / 2; off > 0; off >>= 1) {
    if (t < off) {
      ssum[t] += ssum[t + off];
      ssq[t]  += ssq[t + off];
    }
    __syncthreads();
  }
  const float mean = ssum[0] * (1.f / H_DIM);
  const float var  = ssq[0] * (1.f / H_DIM) - mean * mean;
  Out[m * H_DIM + t] = (r - mean) * rsqrtf(var + 1e-5f) * g[t] + bb[t];
}

// ---------------------------------------------------------------------------
extern "C" void kernel_launch(void* const* d_in, const int* in_sizes, int n_in,
                              void* d_out, int out_size, void* d_ws,
                              size_t ws_size, hipStream_t stream) {
  (void)in_sizes; (void)n_in; (void)out_size; (void)ws_size;

  const float* x     = (const float*)d_in[0];
  const float* enc_w = (const float*)d_in[1];
  const float* enc_b = (const float*)d_in[2];
  const float* lAr   = (const float*)d_in[3];
  const float* Aim   = (const float*)d_in[4];
  const float* Bre   = (const float*)d_in[5];
  const float* Bim   = (const float*)d_in[6];
  const float* Cre   = (const float*)d_in[7];
  const float* Cim   = (const float*)d_in[8];
  const float* ldt   = (const float*)d_in[9];
  const float* Dp    = (const float*)d_in[10];
  const float* ln1g  = (const float*)d_in[11];
  const float* ln1b  = (const float*)d_in[12];
  const float* dec_w = (const float*)d_in[13];
  const float* dec_b = (const float*)d_in[14];
  const float* tcn_w = (const float*)d_in[15];
  const float* tcn_b = (const float*)d_in[16];
  const float* ng    = (const float*)d_in[17];
  const float* nb    = (const float*)d_in[18];
  float* out = (float*)d_out;

  // Scratch layout (bytes): U 33.5MB | Kg 1MB | Yt 33.5MB | Blk 33.5MB
  char*  ws  = (char*)d_ws;
  float* U   = (float*)(ws + 0);
  float* Kg  = (float*)(ws + 33554432);
  float* Yt  = (float*)(ws + 33554432 + 1048576);
  float* Blk = (float*)(ws + 33554432 + 1048576 + 33554432);

  // 1. encoder GEMM: U = x @ enc_w + enc_b        [32768,64]@[64,256]
  k_gemm_bias<DIN><<<4096, 256, 0, stream>>>(x, enc_w, enc_b, U);
  // 2. SSM conv kernel K[h,l]
  k_ssm_kernel<<<dim3(4, 256), 256, 0, stream>>>(lAr, Aim, Bre, Bim, Cre, Cim,
                                                 ldt, Kg);
  // 3. causal Toeplitz conv: Yt[h,l,b]
  k_conv<<<dim3(32, 256), 128, 0, stream>>>(U, Kg, Yt);
  // 4. gelu + skip + LN1 -> Blk
  k_ln1<<<32768, 256, 0, stream>>>(U, Yt, Dp, ln1g, ln1b, Blk);
  // 5. decoder GEMM: out = Blk @ dec_w + dec_b    [32768,256]@[256,256]
  k_gemm_bias<H_DIM><<<4096, 256, 0, stream>>>(Blk, dec_w, dec_b, out);
  // 6. TCN branch accumulated into out
  k_tcn<<<4096, 256, 0, stream>>>(x, tcn_w, tcn_b, out);
  // 7. final LayerNorm in place
  k_lnf<<<32768, 256, 0, stream>>>(out, ng, nb);
}